// TransformerNN_2327872274908
// MI455X (gfx1250) — compile-verified
//
#include <hip/hip_runtime.h>
#include <math.h>

#define NN 30000
#define NE 480000
#define INC 128
#define OUTC 64
#define NH 4
#define EDGED 8
#define HC 256           // NH*OUTC
#define KPAD 132         // LDS column pitch (dwords); 132 % 64 == 4 -> conflict-free b64 reads

typedef float v2f __attribute__((ext_vector_type(2)));
typedef float v8f __attribute__((ext_vector_type(8)));

// ---------------- workspace layout (float offsets) ----------------
#define Q_OFF      ((size_t)0)
#define K_OFF      ((size_t)7680000)      // NN*HC
#define V_OFF      ((size_t)15360000)
#define SKIP_OFF   ((size_t)23040000)     // NN*OUTC
#define G_OFF      ((size_t)24960000)     // NN*NH*EDGED
#define MAX_OFF    ((size_t)25920000)     // NN*NH
#define DEN_OFF    ((size_t)26040000)     // NN*NH
#define ACC_OFF    ((size_t)26160000)     // NN*OUTC (sum over heads of alpha*v; later out_pre in place)
#define T_OFF      ((size_t)28080000)     // NN*NH*EDGED
#define LOGIT_OFF  ((size_t)29040000)     // NE*NH (logits -> ealpha in place)
#define SUMS_OFF   ((size_t)30960000)     // 64
#define SUMSQ_OFF  ((size_t)30960064)     // 64

__device__ inline void atomicMaxFloat(float* addr, float val) {
    // monotone int-ordering trick; addr initialized to -inf bits
    if (val >= 0.0f) atomicMax((int*)addr, __float_as_int(val));
    else             atomicMin((unsigned int*)addr, __float_as_uint(val));
}

// ---------------- init accumulators ----------------
__global__ __launch_bounds__(256)
void init_ws(float* ws) {
    size_t i = (size_t)blockIdx.x * blockDim.x + threadIdx.x;
    if (i < (size_t)NN * NH) {
        ws[MAX_OFF + i] = -INFINITY;
        ws[DEN_OFF + i] = 0.0f;
    }
    if (i < (size_t)NN * OUTC) ws[ACC_OFF + i] = 0.0f;
    if (i < (size_t)NN * NH * EDGED) ws[T_OFF + i] = 0.0f;
    if (i < 128) ws[SUMS_OFF + i] = 0.0f;   // covers sums[64]+sumsq[64] (contiguous)
}

// ---------------- fp32 WMMA GEMM: Y[M,Ncols] = X[M,Kdim] @ W[Kdim,Ncols] + bias ----------------
// Block = 8 waves. Block tile: 128(M) x 64(N). W[:,64-slice] staged TRANSPOSED in LDS
// (sW[col][k], pitch KPAD) so each B operand is one contiguous ds_load_b64.
// Each wave32 computes a 16(M) x 64(N) strip: 4 accumulators, A loaded once per K-step
// and reused for 4 x V_WMMA_F32_16X16X4_F32.
__global__ __launch_bounds__(256)
void gemm_wmma_f32(const float* __restrict__ X, const float* __restrict__ W,
                   const float* __restrict__ bias, float* __restrict__ Y,
                   int M, int Kdim, int Ncols)
{
    __shared__ float sW[64 * KPAD];             // sW[col * KPAD + k], col in [0,64)
    int tid = threadIdx.y * 32 + threadIdx.x;
    int n0  = blockIdx.y * 64;

    // cooperative transpose-stage: W[k, n0+c] -> sW[c][k]
    for (int idx = tid; idx < Kdim * 16; idx += 256) {
        int kk = idx >> 4;
        int c4 = (idx & 15) * 4;
        float4 w = *(const float4*)&W[(size_t)kk * Ncols + n0 + c4];
        sW[(c4 + 0) * KPAD + kk] = w.x;
        sW[(c4 + 1) * KPAD + kk] = w.y;
        sW[(c4 + 2) * KPAD + kk] = w.z;
        sW[(c4 + 3) * KPAD + kk] = w.w;
    }
    __syncthreads();

    int lane  = threadIdx.x;                    // 0..31
    int mtile = blockIdx.x * 8 + threadIdx.y;
    if (mtile * 16 >= M) return;                // uniform per wave; after the barrier
    int m0   = mtile * 16;
    int r    = lane & 15;                       // A row (M) and B/D column (N) within tile
    int koff = (lane >> 4) * 2;                 // K sub-offset: 0 or 2

    v8f acc0 = {}, acc1 = {}, acc2 = {}, acc3 = {};
    const float* xrow = X + (size_t)(m0 + r) * Kdim;
    const float* bcol = sW + r * KPAD;          // lane's base column; tiles at +16/32/48 cols
    for (int k = 0; k < Kdim; k += 4) {
        v2f a  = *(const v2f*)(xrow + k + koff);            // A[M, k+koff .. +1]
        v2f b0 = *(const v2f*)(bcol + (k + koff));          // W[k+koff..+1, n0+ 0+r]
        v2f b1 = *(const v2f*)(bcol + 16 * KPAD + (k + koff));
        v2f b2 = *(const v2f*)(bcol + 32 * KPAD + (k + koff));
        v2f b3 = *(const v2f*)(bcol + 48 * KPAD + (k + koff));
        acc0 = __builtin_amdgcn_wmma_f32_16x16x4_f32(false, a, false, b0, (short)0, acc0, false, false);
        acc1 = __builtin_amdgcn_wmma_f32_16x16x4_f32(false, a, false, b1, (short)0, acc1, false, false);
        acc2 = __builtin_amdgcn_wmma_f32_16x16x4_f32(false, a, false, b2, (short)0, acc2, false, false);
        acc3 = __builtin_amdgcn_wmma_f32_16x16x4_f32(false, a, false, b3, (short)0, acc3, false, false);
    }

    int mb = (lane >> 4) * 8;                   // C/D: VGPR i -> M = i + 8*(lane>>4)
    float bv0 = bias ? bias[n0 +  0 + r] : 0.0f;
    float bv1 = bias ? bias[n0 + 16 + r] : 0.0f;
    float bv2 = bias ? bias[n0 + 32 + r] : 0.0f;
    float bv3 = bias ? bias[n0 + 48 + r] : 0.0f;
    #pragma unroll
    for (int i = 0; i < 8; ++i) {
        size_t row = (size_t)(m0 + mb + i) * Ncols + n0 + r;
        Y[row +  0] = acc0[i] + bv0;
        Y[row + 16] = acc1[i] + bv1;
        Y[row + 32] = acc2[i] + bv2;
        Y[row + 48] = acc3[i] + bv3;
    }
}

// ---------------- g[n,h,d] = sum_c q[n,h,c] * We[d, h*64+c] ----------------
__global__ __launch_bounds__(256)
void compute_g(const float* __restrict__ q, const float* __restrict__ We,
               float* __restrict__ gbuf)
{
    int i = blockIdx.x * blockDim.x + threadIdx.x;
    if (i >= NN * NH * EDGED) return;
    int n = i >> 5;
    int h = (i >> 3) & 3;
    int d = i & 7;
    const float* qp = q  + (size_t)n * HC + h * OUTC;
    const float* wp = We + (size_t)d * HC + h * OUTC;
    float s = 0.0f;
    #pragma unroll
    for (int c = 0; c < OUTC; c += 4) {
        float4 qa = *(const float4*)(qp + c);
        float4 wa = *(const float4*)(wp + c);
        s += qa.x * wa.x + qa.y * wa.y + qa.z * wa.z + qa.w * wa.w;
    }
    gbuf[i] = s;
}

// ---------------- edge pass 1: logits + scatter max (one wave per edge) ----------------
__global__ __launch_bounds__(256)
void edge_logits(const float* __restrict__ q, const float* __restrict__ k,
                 const float* __restrict__ g, const int* __restrict__ ei,
                 const float* __restrict__ ea,
                 float* __restrict__ logit, float* __restrict__ maxa)
{
    int tid = blockIdx.x * blockDim.x + threadIdx.x;
    int e = tid >> 5;
    if (e >= NE) return;
    int lane = tid & 31;
    int h = lane >> 3, j = lane & 7;
    int src = ei[e], dst = ei[NE + e];
    const float* qp = q + (size_t)dst * HC + h * OUTC + j * 8;
    const float* kp = k + (size_t)src * HC + h * OUTC + j * 8;
    float4 q0 = *(const float4*)(qp);
    float4 q1 = *(const float4*)(qp + 4);
    float4 k0 = *(const float4*)(kp);
    float4 k1 = *(const float4*)(kp + 4);
    float p = q0.x * k0.x + q0.y * k0.y + q0.z * k0.z + q0.w * k0.w
            + q1.x * k1.x + q1.y * k1.y + q1.z * k1.z + q1.w * k1.w;
    // edge-feature logit term folded into same reduction: ea[e,j]*g[dst,h,j]
    p += ea[(size_t)e * EDGED + j] * g[(size_t)dst * (NH * EDGED) + h * EDGED + j];
    p += __shfl_down(p, 4, 8);
    p += __shfl_down(p, 2, 8);
    p += __shfl_down(p, 1, 8);
    if (j == 0) {
        float lg = p * 0.125f;              // / sqrt(OUTC)
        logit[(size_t)e * NH + h] = lg;
        atomicMaxFloat(&maxa[(size_t)dst * NH + h], lg);
    }
}

// ---------------- edge pass 2: exp(logit - max) + scatter sum ----------------
__global__ __launch_bounds__(256)
void edge_softmax_num(const int* __restrict__ ei, float* __restrict__ logit,
                      const float* __restrict__ maxa, float* __restrict__ denom)
{
    int i = blockIdx.x * blockDim.x + threadIdx.x;
    if (i >= NE * NH) return;
    int e = i >> 2, h = i & 3;
    int dst = ei[NE + e];
    float m = maxa[(size_t)dst * NH + h];
    if (!isfinite(m)) m = 0.0f;
    float ez = __expf(logit[i] - m);
    logit[i] = ez;                          // becomes ealpha in place
    atomicAdd(&denom[(size_t)dst * NH + h], ez);
}

// ---------------- edge pass 3: weighted aggregation (64 threads per edge) ----------------
__global__ __launch_bounds__(256)
void edge_aggregate(const int* __restrict__ ei, const float* __restrict__ ea,
                    const float* __restrict__ v, const float* __restrict__ ealpha,
                    const float* __restrict__ denom,
                    float* __restrict__ acc, float* __restrict__ tacc)
{
    __shared__ float s_a[4][NH];
    int slot = threadIdx.x >> 6;            // edge slot in block: 0..3
    int c    = threadIdx.x & 63;            // output channel
    int e = blockIdx.x * 4 + slot;
    bool valid = (e < NE);
    int src = 0, dst = 0;
    if (valid) { src = ei[e]; dst = ei[NE + e]; }
    if (valid && c < NH) {
        float d = denom[(size_t)dst * NH + c];
        s_a[slot][c] = ealpha[(size_t)e * NH + c] / (d + 1e-16f);
    }
    __syncthreads();
    if (!valid) return;
    float a0 = s_a[slot][0], a1 = s_a[slot][1], a2 = s_a[slot][2], a3 = s_a[slot][3];
    const float* vp = v + (size_t)src * HC;
    // head-mean folded later; sum heads now -> one atomic per channel
    float val = a0 * vp[c] + a1 * vp[64 + c] + a2 * vp[128 + c] + a3 * vp[192 + c];
    atomicAdd(&acc[(size_t)dst * OUTC + c], val);
    if (c < NH * EDGED) {                   // t[dst,h,d] += alpha[h]*ea[e,d]
        int h = c >> 3, d = c & 7;
        atomicAdd(&tacc[(size_t)dst * (NH * EDGED) + c],
                  s_a[slot][h] * ea[(size_t)e * EDGED + d]);
    }
}

// ---------------- finalize pre-BN + batch statistics ----------------
__global__ __launch_bounds__(256)
void finalize_stats(const float* __restrict__ tacc, const float* __restrict__ We,
                    const float* __restrict__ skip, float* __restrict__ acc,
                    float* __restrict__ sums, float* __restrict__ sumsq)
{
    int c = threadIdx.x & 63;
    int r = threadIdx.x >> 6;               // 0..3
    float lsum = 0.0f, lsq = 0.0f;
    for (int n = blockIdx.x * 4 + r; n < NN; n += gridDim.x * 4) {
        float ev = 0.0f;
        #pragma unroll
        for (int h = 0; h < NH; ++h)
            #pragma unroll
            for (int d = 0; d < EDGED; ++d)
                ev += tacc[(size_t)n * (NH * EDGED) + h * EDGED + d] *
                      We[(size_t)d * HC + h * OUTC + c];
        float val = 0.25f * (acc[(size_t)n * OUTC + c] + ev)   // mean over heads
                  + skip[(size_t)n * OUTC + c];
        acc[(size_t)n * OUTC + c] = val;                        // out_pre in place
        lsum += val; lsq += val * val;
    }
    __shared__ float ssum[256], ssq[256];
    ssum[threadIdx.x] = lsum; ssq[threadIdx.x] = lsq;
    __syncthreads();
    if (r == 0) {
        atomicAdd(&sums[c],  ssum[c] + ssum[64 + c] + ssum[128 + c] + ssum[192 + c]);
        atomicAdd(&sumsq[c], ssq[c]  + ssq[64 + c]  + ssq[128 + c]  + ssq[192 + c]);
    }
}

// ---------------- BN apply ----------------
__global__ __launch_bounds__(256)
void bn_apply(const float* __restrict__ pre, const float* __restrict__ sums,
              const float* __restrict__ sumsq, const float* __restrict__ gamma,
              const float* __restrict__ beta, float* __restrict__ out)
{
    int i = blockIdx.x * blockDim.x + threadIdx.x;
    if (i >= NN * OUTC) return;
    int c = i & 63;
    float mu  = sums[c]  * (1.0f / NN);
    float var = sumsq[c] * (1.0f / NN) - mu * mu;
    float inv = rsqrtf(var + 1e-5f);
    out[i] = gamma[c] * (pre[i] - mu) * inv + beta[c];
}

extern "C" void kernel_launch(void* const* d_in, const int* in_sizes, int n_in,
                              void* d_out, int out_size, void* d_ws, size_t ws_size,
                              hipStream_t stream) {
    const float* x     = (const float*)d_in[0];
    const int*   ei    = (const int*)  d_in[1];
    const float* ea    = (const float*)d_in[2];
    const float* Wq    = (const float*)d_in[3];
    const float* bq    = (const float*)d_in[4];
    const float* Wk    = (const float*)d_in[5];
    const float* bk    = (const float*)d_in[6];
    const float* Wv    = (const float*)d_in[7];
    const float* bv    = (const float*)d_in[8];
    const float* We    = (const float*)d_in[9];
    const float* Wskip = (const float*)d_in[10];
    const float* bskip = (const float*)d_in[11];
    const float* gamma = (const float*)d_in[12];
    const float* beta  = (const float*)d_in[13];
    float* ws  = (float*)d_ws;
    float* out = (float*)d_out;

    // 0) init accumulators
    init_ws<<<(NN * OUTC + 255) / 256, 256, 0, stream>>>(ws);

    // 1) WMMA GEMMs: Q,K,V [NN,256] and skip [NN,64]. 30000 = 1875 * 16 (no tail).
    dim3 blk(32, 8);
    dim3 grdQ((1875 + 7) / 8, HC / 64);     // block tile 128x64
    dim3 grdS((1875 + 7) / 8, OUTC / 64);
    gemm_wmma_f32<<<grdQ, blk, 0, stream>>>(x, Wq, bq, ws + Q_OFF, NN, INC, HC);
    gemm_wmma_f32<<<grdQ, blk, 0, stream>>>(x, Wk, bk, ws + K_OFF, NN, INC, HC);
    gemm_wmma_f32<<<grdQ, blk, 0, stream>>>(x, Wv, bv, ws + V_OFF, NN, INC, HC);
    gemm_wmma_f32<<<grdS, blk, 0, stream>>>(x, Wskip, bskip, ws + SKIP_OFF, NN, INC, OUTC);

    // 2) g = per-node projection of q onto We rows (for edge logit term)
    compute_g<<<(NN * NH * EDGED + 255) / 256, 256, 0, stream>>>(ws + Q_OFF, We, ws + G_OFF);

    // 3) edge logits + scatter-max (one wave32 per edge)
    edge_logits<<<(NE * 32 + 255) / 256, 256, 0, stream>>>(
        ws + Q_OFF, ws + K_OFF, ws + G_OFF, ei, ea, ws + LOGIT_OFF, ws + MAX_OFF);

    // 4) exp + scatter-sum denominator
    edge_softmax_num<<<(NE * NH + 255) / 256, 256, 0, stream>>>(
        ei, ws + LOGIT_OFF, ws + MAX_OFF, ws + DEN_OFF);

    // 5) weighted aggregation
    edge_aggregate<<<(NE + 3) / 4, 256, 0, stream>>>(
        ei, ea, ws + V_OFF, ws + LOGIT_OFF, ws + DEN_OFF, ws + ACC_OFF, ws + T_OFF);

    // 6) finalize (head-mean + edge-feature reconstruction + skip) + BN stats
    finalize_stats<<<120, 256, 0, stream>>>(
        ws + T_OFF, We, ws + SKIP_OFF, ws + ACC_OFF, ws + SUMS_OFF, ws + SUMSQ_OFF);

    // 7) BN apply
    bn_apply<<<(NN * OUTC + 255) / 256, 256, 0, stream>>>(
        ws + ACC_OFF, ws + SUMS_OFF, ws + SUMSQ_OFF, gamma, beta, out);
}